// CRFModule_83442624627194
// MI455X (gfx1250) — compile-verified
//
#include <hip/hip_runtime.h>
#include <hip/hip_bf16.h>
#include <math.h>

#define TAG  128
#define BATCH 128
#define TLEN 1024
#define NB   16      // batches per workgroup == WMMA N
#define NTHREADS 256 // 8 waves of 32

typedef __attribute__((ext_vector_type(2))) float v2f;
typedef __attribute__((ext_vector_type(4))) float v4f;
typedef __attribute__((ext_vector_type(8))) float v8f;

// ---------------------------------------------------------------------------
// Forward algorithm: one workgroup handles NB=16 batch chains for all T steps.
// Per step:  S = exp(trans) (128x128, f32, A-frags resident in regs)
//              x  P (128x16, LDS, B-fragment-swizzled)
// via 8 waves x 32 v_wmma_f32_16x16x4_f32 (two independent accumulator chains).
// feats/mask for step t are loaded at the top of the loop so the global-memory
// latency overlaps the exp/shuffle stage and the WMMA chain.
// ---------------------------------------------------------------------------
__global__ __launch_bounds__(NTHREADS) void crf_forward_kernel(
    const float* __restrict__ feats,          // [B][T][TAG]
    const float* __restrict__ trans,          // [TAG][TAG]  trans[i][j]
    const float* __restrict__ startT,         // [TAG]
    const float* __restrict__ stopT,          // [TAG]
    const unsigned char* __restrict__ mask,   // [B][T]
    float* __restrict__ fwd_out)              // [B]
{
    __shared__ float fv[NB][TAG + 1];   // padded: kill bank conflicts
    __shared__ v2f   Pl[TAG / 2][NB];   // p in B-fragment layout: Pl[j>>1][b]
    __shared__ float Mb[NB];            // per-batch max

    const int tid  = threadIdx.x;
    const int wave = tid >> 5;          // 0..7  -> owns S rows [16w,16w+16)
    const int lane = tid & 31;
    const int b0   = blockIdx.x * NB;   // global batch base

    const int bcol  = lane & 15;        // WMMA N column == local batch
    const int bhalf = lane >> 4;        // 0: K rows {4c,4c+1} / M rows r..r+7
                                        // 1: K rows {4c+2,4c+3} / M rows r+8..

    // ---- preload A fragments: E[i][j] = exp(trans[i][j]), rows 16w..16w+15 ----
    // A 16x4 f32 layout: lanes 0-15 hold (K=4c,4c+1), lanes 16-31 (K=4c+2,4c+3)
    v2f Afrag[32];
    {
        const int m     = wave * 16 + (lane & 15);
        const int khalf = bhalf * 2;
        const float* tr = trans + (size_t)m * TAG;
        for (int c = 0; c < 32; ++c) {
            const int k = c * 4 + khalf;
            Afrag[c].x = __expf(tr[k]);
            Afrag[c].y = __expf(tr[k + 1]);
        }
    }

    // ---- init fv[b][j] = start[j] + feats[b][0][j] ----
    {
        const int b  = tid >> 4;            // 0..15 local batch
        const int j0 = (tid & 15) * 8;
        const float* f0 = feats + (size_t)(b0 + b) * TLEN * TAG;
        for (int r = 0; r < 8; ++r)
            fv[b][j0 + r] = startT[j0 + r] + f0[j0 + r];
    }
    __syncthreads();

    // stage-3 mapping constants (C/D layout: reg r, lanes 0-15 -> M=r, N=lane;
    // lanes 16-31 -> M=r+8, N=lane-16)
    const int bb = bcol;                           // local batch of this lane
    const int i0 = wave * 16 + bhalf * 8;          // first S row this lane holds
    const float* fbase = feats + ((size_t)(b0 + bb) * TLEN) * TAG + i0;
    const unsigned char* mbase = mask + (size_t)(b0 + bb) * TLEN;

    for (int t = 1; t < TLEN; ++t) {
        // ---- issue this step's global loads FIRST: they depend on nothing
        //      computed in this iteration, so their latency hides under
        //      stage 1 and the WMMA chain ----
        const float* fptr = fbase + (size_t)t * TAG;
        const v4f fa = ((const v4f*)fptr)[0];
        const v4f fc = ((const v4f*)fptr)[1];
        const unsigned char mk = mbase[t];
        __builtin_prefetch(fptr + TAG, 0, 0);      // warm L2/L0 for t+1

        // ---- stage 1: per-batch max + p = exp(fv - M), swizzled into Pl ----
        const int b  = tid >> 4;
        const int j0 = (tid & 15) * 8;
        float loc[8];
        float m = -INFINITY;
        for (int r = 0; r < 8; ++r) {
            loc[r] = fv[b][j0 + r];
            m = fmaxf(m, loc[r]);
        }
        // reduce max across the 16 threads of this batch (a half-wave)
        for (int off = 8; off; off >>= 1)
            m = fmaxf(m, __shfl_xor(m, off, 16));
        if ((tid & 15) == 0) Mb[b] = m;
        for (int q = 0; q < 4; ++q) {
            v2f pv;
            pv.x = __expf(loc[2 * q]     - m);
            pv.y = __expf(loc[2 * q + 1] - m);
            Pl[(j0 >> 1) + q][b] = pv;      // Pl[jpair][b]
        }
        __syncthreads();

        const float Mbb = Mb[bb];           // read early, covered by WMMA chain

        // ---- stage 2: B fragments from LDS (compiler software-pipelines the
        //      ds_load_2addr_b64s ahead of the consuming WMMAs), two
        //      independent WMMA accumulator chains ----
        v2f Bfrag[32];
        for (int c = 0; c < 32; ++c)
            Bfrag[c] = Pl[c * 2 + bhalf][bcol];

        v8f acc0 = {0.f, 0.f, 0.f, 0.f, 0.f, 0.f, 0.f, 0.f};
        v8f acc1 = {0.f, 0.f, 0.f, 0.f, 0.f, 0.f, 0.f, 0.f};
        for (int c = 0; c < 32; c += 2) {
            acc0 = __builtin_amdgcn_wmma_f32_16x16x4_f32(
                       false, Afrag[c],     false, Bfrag[c],     (short)0, acc0,
                       false, false);
            acc1 = __builtin_amdgcn_wmma_f32_16x16x4_f32(
                       false, Afrag[c + 1], false, Bfrag[c + 1], (short)0, acc1,
                       false, false);
        }

        // ---- stage 3: fv_new[b][i] = feat + M_b + log(S[i][b]) ----
        float nf[8];
        for (int r = 0; r < 4; ++r)
            nf[r] = fa[r] + Mbb + __logf(acc0[r] + acc1[r]);
        for (int r = 4; r < 8; ++r)
            nf[r] = fc[r - 4] + Mbb + __logf(acc0[r] + acc1[r]);
        for (int r = 0; r < 8; ++r) {
            const float oldv = fv[bb][i0 + r];      // this lane is sole owner
            fv[bb][i0 + r] = mk ? nf[r] : oldv;
        }
        __syncthreads();
    }

    // ---- final: fwd[b] = logsumexp_i(fv[b][i] + stop[i]) ----
    {
        const int b  = tid >> 4;
        const int j0 = (tid & 15) * 8;
        float v[8];
        float m = -INFINITY;
        for (int r = 0; r < 8; ++r) {
            v[r] = fv[b][j0 + r] + stopT[j0 + r];
            m = fmaxf(m, v[r]);
        }
        for (int off = 8; off; off >>= 1)
            m = fmaxf(m, __shfl_xor(m, off, 16));
        float s = 0.f;
        for (int r = 0; r < 8; ++r) s += __expf(v[r] - m);
        for (int off = 8; off; off >>= 1)
            s += __shfl_xor(s, off, 16);
        if ((tid & 15) == 0) fwd_out[b0 + b] = m + __logf(s);
    }
}

// ---------------------------------------------------------------------------
// Gold path score: one wave per batch, lanes stride over T, shfl reduction.
// ---------------------------------------------------------------------------
__global__ __launch_bounds__(256) void crf_gold_kernel(
    const float* __restrict__ feats, const float* __restrict__ trans,
    const float* __restrict__ startT, const float* __restrict__ stopT,
    const long long* __restrict__ tags, const unsigned char* __restrict__ mask,
    float* __restrict__ gold_out)
{
    const int wave = threadIdx.x >> 5;
    const int lane = threadIdx.x & 31;
    const int b = blockIdx.x * 8 + wave;
    if (b >= BATCH) return;

    const long long*     tg = tags  + (size_t)b * TLEN;
    const unsigned char* mk = mask  + (size_t)b * TLEN;
    const float*         f  = feats + (size_t)b * TLEN * TAG;

    float s = 0.f;
    int cnt = 0;
    for (int t = lane; t < TLEN; t += 32) {
        const int cur = (int)tg[t];
        cnt += mk[t] ? 1 : 0;
        if (t >= 1 && mk[t]) {
            const int prev = (int)tg[t - 1];
            s += trans[(size_t)cur * TAG + prev] + f[(size_t)t * TAG + cur];
        }
    }
    for (int off = 16; off; off >>= 1) {
        s   += __shfl_xor(s, off, 32);
        cnt += __shfl_xor(cnt, off, 32);
    }
    if (lane == 0) {
        const int t0 = (int)tg[0];
        float sc = s + startT[t0] + f[t0];          // + emit[b,0]
        const int last_idx = cnt - 1;               // mask.sum() - 1
        sc += stopT[(int)tg[last_idx]];
        gold_out[b] = sc;
    }
}

// ---------------------------------------------------------------------------
// out = mean(fwd - gold)
// ---------------------------------------------------------------------------
__global__ __launch_bounds__(BATCH) void crf_combine_kernel(
    const float* __restrict__ fwd, const float* __restrict__ gold,
    float* __restrict__ out)
{
    const int tid = threadIdx.x;
    __shared__ float red[BATCH / 32];
    float v = fwd[tid] - gold[tid];
    for (int off = 16; off; off >>= 1) v += __shfl_xor(v, off, 32);
    if ((tid & 31) == 0) red[tid >> 5] = v;
    __syncthreads();
    if (tid == 0) {
        float s = 0.f;
        for (int w = 0; w < BATCH / 32; ++w) s += red[w];
        out[0] = s / (float)BATCH;
    }
}

extern "C" void kernel_launch(void* const* d_in, const int* in_sizes, int n_in,
                              void* d_out, int out_size, void* d_ws, size_t ws_size,
                              hipStream_t stream) {
    const float*         feats  = (const float*)d_in[0];
    const float*         trans  = (const float*)d_in[1];
    const float*         startT = (const float*)d_in[2];
    const float*         stopT  = (const float*)d_in[3];
    const long long*     tags   = (const long long*)d_in[4];
    const unsigned char* mask   = (const unsigned char*)d_in[5];

    float* fwd  = (float*)d_ws;          // [BATCH]
    float* gold = fwd + BATCH;           // [BATCH]

    crf_gold_kernel<<<BATCH / 8, 256, 0, stream>>>(feats, trans, startT, stopT,
                                                   tags, mask, gold);
    crf_forward_kernel<<<BATCH / NB, NTHREADS, 0, stream>>>(feats, trans, startT,
                                                            stopT, mask, fwd);
    crf_combine_kernel<<<1, BATCH, 0, stream>>>(fwd, gold, (float*)d_out);
}